// Model_24842090840521
// MI455X (gfx1250) — compile-verified
//
#include <hip/hip_runtime.h>

// MI455X / gfx1250, wave32. Fused: 1x1-conv GEMM (fp32 WMMA 16x16x4) + switch
// matvec + relu + per-batch top-16 + gather/scatter, one workgroup per batch.
//
// B operand is staged in LDS in WMMA-ready per-lane float2 layout so every
// B fetch is one conflict-free ds_load_b64 into an even-aligned VGPR pair
// (no register-shuffle movs feeding v_wmma).

typedef __attribute__((ext_vector_type(2))) float v2f;
typedef __attribute__((ext_vector_type(8))) float v8f;

#define NB    256   // batches
#define CIN   1024  // encoder channels (GEMM K)
#define TT    128   // time frames (GEMM M per batch)
#define CTX   64    // context dim (GEMM N)
#define NEV   16    // top-k
#define KC    64    // K-chunk staged in LDS
#define NG    (KC / 4)          // K-groups per chunk (one WMMA K-step each)
#define PITCH 65    // ev_lds row pitch (bank-conflict pad)

__global__ __launch_bounds__(256)
void fused_events_kernel(const float* __restrict__ encoded,   // [B, CIN, TT]
                         const float* __restrict__ w_vec,     // [CTX, CIN]
                         const float* __restrict__ b_vec,     // [CTX]
                         const float* __restrict__ w_switch,  // [CIN]
                         const float* __restrict__ b_switch,  // [1]
                         float* __restrict__ out_vecs,        // [B, NEV, CTX]
                         float* __restrict__ out_sched)       // [B, NEV, TT]
{
    // bop[g][dt][lane] = float2 B-fragment for WMMA: k = 4g + 2*(lane>>4),
    // d = 16*dt + (lane&15); value = { w_vec[d][c0+k], w_vec[d][c0+k+1] }
    __shared__ float bop[NG * 4 * 32 * 2];     // 16 KB, WMMA-ready B chunks
    __shared__ float ws_lds[KC];               // w_switch chunk
    __shared__ float bvec_lds[CTX];
    __shared__ float ev_lds[TT * PITCH];       // event_vecs for this batch: [t][d]
    __shared__ float attn_lds[TT];
    __shared__ float vals_lds[NEV];
    __shared__ int   idx_lds[NEV];

    const int tid  = threadIdx.x;
    const int lane = tid & 31;
    const int wave = tid >> 5;          // 0..7 -> t-tile
    const int half = lane >> 4;         // K-pair selector in A layout
    const int tcol = lane & 15;         // M (t) for A, N (d) for B/D
    const int b    = blockIdx.x;
    const int t0   = wave * 16;

    const float* encB = encoded + (size_t)b * CIN * TT;

    if (tid < CTX) bvec_lds[tid] = b_vec[tid];

    v8f acc0 = {}, acc1 = {}, acc2 = {}, acc3 = {};
    float sw = 0.0f;

    for (int c0 = 0; c0 < CIN; c0 += KC) {
        __syncthreads();   // previous chunk fully consumed
        // ---- stage WMMA-ready B chunk: thread -> one d, 16 contiguous c's ----
        {
            const int d  = tid >> 2;          // 0..63
            const int cq = (tid & 3) * 16;    // 0,16,32,48
            const int dt = d >> 4;            // d-tile 0..3
            const int dl = d & 15;            // lane column within tile
            const float* src = w_vec + (size_t)d * CIN + c0 + cq;
            #pragma unroll
            for (int j = 0; j < 16; j += 4) {
                float4 w4 = *(const float4*)(src + j);
                const int cc = cq + j;        // local k of w4.x (multiple of 4)
                // k = cc+q -> g = (cc+q)>>2, half = ((cc+q)>>1)&1, lo = (cc+q)&1
                const int base = ((cc >> 2) * 4 + dt) * 64 + dl * 2;
                bop[base +  0] = w4.x;        // q=0: half 0, lo 0
                bop[base +  1] = w4.y;        // q=1: half 0, lo 1
                bop[base + 32] = w4.z;        // q=2: half 1, lo 0
                bop[base + 33] = w4.w;        // q=3: half 1, lo 1
            }
            if (c0 + KC < CIN)   // warm L2/GL2 for the next chunk
                __builtin_prefetch(src + KC, 0, 0);
        }
        if (tid < KC) ws_lds[tid] = w_switch[c0 + tid];
        __syncthreads();

        const v2f* bopv = (const v2f*)bop;

        // ---- K loop over this chunk: 4 K-values per WMMA step ----
        #pragma unroll 4
        for (int g = 0; g < NG; g++) {
            // A 16x4: lanes 0-15 hold K = 4g,4g+1 ; lanes 16-31 hold 4g+2,4g+3
            const int cA = c0 + 4 * g + 2 * half;
            v2f a;
            a.x = encB[(size_t)(cA + 0) * TT + t0 + tcol];
            a.y = encB[(size_t)(cA + 1) * TT + t0 + tcol];

            // switch matvec rides along (same operands, per-lane partial)
            sw = fmaf(a.x, ws_lds[4 * g + 2 * half + 0], sw);
            sw = fmaf(a.y, ws_lds[4 * g + 2 * half + 1], sw);

            // B fragments: one ds_load_b64 per d-tile, lane-linear, conflict-free
            const int bbase = (g * 4) * 32 + lane;
            v2f b0 = bopv[bbase +  0];
            v2f b1 = bopv[bbase + 32];
            v2f b2 = bopv[bbase + 64];
            v2f b3 = bopv[bbase + 96];

            acc0 = __builtin_amdgcn_wmma_f32_16x16x4_f32(false, a, false, b0, (short)0, acc0, false, false);
            acc1 = __builtin_amdgcn_wmma_f32_16x16x4_f32(false, a, false, b1, (short)0, acc1, false, false);
            acc2 = __builtin_amdgcn_wmma_f32_16x16x4_f32(false, a, false, b2, (short)0, acc2, false, false);
            acc3 = __builtin_amdgcn_wmma_f32_16x16x4_f32(false, a, false, b3, (short)0, acc3, false, false);
        }
    }

    // ---- spill event_vecs tiles (+bias) to LDS; D: VGPR r -> M = r + 8*half ----
    #pragma unroll
    for (int r = 0; r < 8; r++) {
        const int t = t0 + r + 8 * half;
        ev_lds[t * PITCH +      tcol] = acc0[r] + bvec_lds[     tcol];
        ev_lds[t * PITCH + 16 + tcol] = acc1[r] + bvec_lds[16 + tcol];
        ev_lds[t * PITCH + 32 + tcol] = acc2[r] + bvec_lds[32 + tcol];
        ev_lds[t * PITCH + 48 + tcol] = acc3[r] + bvec_lds[48 + tcol];
    }

    // ---- finalize switch: combine lane halves (wave32 xor-16), bias, relu ----
    sw += __shfl_xor(sw, 16, 32);
    if (half == 0) {
        float aa = sw + b_switch[0];
        attn_lds[t0 + tcol] = aa > 0.0f ? aa : 0.0f;
    }
    __syncthreads();

    // ---- top-16 over TT=128 (wave 0, register-resident iterative argmax) ----
    if (wave == 0) {
        float v0 = attn_lds[lane * 4 + 0];
        float v1 = attn_lds[lane * 4 + 1];
        float v2 = attn_lds[lane * 4 + 2];
        float v3 = attn_lds[lane * 4 + 3];
        for (int j = 0; j < NEV; j++) {
            float bv = v0; int bi = lane * 4;
            if (v1 > bv) { bv = v1; bi = lane * 4 + 1; }
            if (v2 > bv) { bv = v2; bi = lane * 4 + 2; }
            if (v3 > bv) { bv = v3; bi = lane * 4 + 3; }
            #pragma unroll
            for (int off = 16; off >= 1; off >>= 1) {
                float ov = __shfl_xor(bv, off, 32);
                int   oi = __shfl_xor(bi, off, 32);
                // prefer larger value; on tie, lower index (matches lax.top_k)
                if (ov > bv || (ov == bv && oi < bi)) { bv = ov; bi = oi; }
            }
            if ((bi >> 2) == lane) {              // knock out the winner locally
                if      ((bi & 3) == 0) v0 = -1.0f;
                else if ((bi & 3) == 1) v1 = -1.0f;
                else if ((bi & 3) == 2) v2 = -1.0f;
                else                    v3 = -1.0f;
            }
            if (lane == 0) { idx_lds[j] = bi; vals_lds[j] = bv; }
        }
    }
    __syncthreads();

    // ---- gather vecs[b,j,:] = ev_lds[idx[j],:]  (16x64 floats, float4 stores) ----
    {
        const int j  = tid >> 4;            // 0..15
        const int d0 = (tid & 15) * 4;      // 0..60
        const int t  = idx_lds[j];
        float4 v;
        v.x = ev_lds[t * PITCH + d0 + 0];
        v.y = ev_lds[t * PITCH + d0 + 1];
        v.z = ev_lds[t * PITCH + d0 + 2];
        v.w = ev_lds[t * PITCH + d0 + 3];
        *(float4*)(out_vecs + ((size_t)b * NEV + j) * CTX + d0) = v;
    }

    // ---- scheduling[b,j,t] = (t == idx[j]) ? vals[j] : 0  (dense write) ----
    {
        const int j   = tid >> 4;
        const int tb  = (tid & 15) * 8;     // 8 consecutive t's per thread
        const int ti  = idx_lds[j];
        const float tv = vals_lds[j];
        float* dst = out_sched + ((size_t)b * NEV + j) * TT + tb;
        #pragma unroll
        for (int q = 0; q < 8; q++)
            dst[q] = (tb + q == ti) ? tv : 0.0f;
    }
}

extern "C" void kernel_launch(void* const* d_in, const int* in_sizes, int n_in,
                              void* d_out, int out_size, void* d_ws, size_t ws_size,
                              hipStream_t stream) {
    (void)in_sizes; (void)n_in; (void)out_size; (void)d_ws; (void)ws_size;
    const float* encoded  = (const float*)d_in[0];
    const float* w_vec    = (const float*)d_in[1];
    const float* b_vec    = (const float*)d_in[2];
    const float* w_switch = (const float*)d_in[3];
    const float* b_switch = (const float*)d_in[4];
    float* out       = (float*)d_out;
    float* out_vecs  = out;                                   // [B,16,64]
    float* out_sched = out + (size_t)NB * NEV * CTX;          // [B,16,128]
    hipLaunchKernelGGL(fused_events_kernel, dim3(NB), dim3(256), 0, stream,
                       encoded, w_vec, b_vec, w_switch, b_switch,
                       out_vecs, out_sched);
}